// QuantumLayer_87634512708276
// MI455X (gfx1250) — compile-verified
//
#include <hip/hip_runtime.h>

// ---------------------------------------------------------------------------
// 16-qubit RY/CNOT state-vector simulator for MI455X (gfx1250).
//   - 1 workgroup per batch element; full 2^16-float state lives in 256KB LDS
//     (CDNA5 WGP has 320KB LDS -> whole state resident, zero HBM round trips).
//   - Each RY round = 4 nibble-wise 16x16 unitaries applied with
//     v_wmma_f32_16x16x4_f32 (K=16 done as 4 chained WMMAs).
//   - The 16-CNOT ring chain per layer collapses analytically to ONE
//     XOR-linear index permutation (prefix-xor), done as a register-staged
//     LDS gather/scatter.
//   - Tile loop is scalarized (readfirstlane + constant trip count) so WMMA
//     executes under uniform control flow with EXEC all-ones, as the ISA
//     requires.
// ---------------------------------------------------------------------------

typedef float v2f __attribute__((ext_vector_type(2)));
typedef float v8f __attribute__((ext_vector_type(8)));

#define NQ      16
#define NLAYERS 8
#define NSTATE  65536          // 2^16 amplitudes
#define TPB     1024           // 32 wave32s
#define PER_T   (NSTATE / TPB) // 64 amplitudes per thread

// Inverse of the composed CNOT chain: CNOT(0,1)..CNOT(14,15), CNOT(15,0).
// Forward map: y_q = b0^..^b_q (q>=1), y_0 = b1^..^b15.
// Inverse:     b_q = y_q ^ y_{q-1} (q>=2), b0 = y0^y15, b1 = y1^b0.
__device__ __forceinline__ int cnot_chain_src(int y) {
  int x   = (y ^ (y << 1)) & 0xFFFC;   // bits 2..15
  int y0  = y & 1;
  int y1  = (y >> 1) & 1;
  int y15 = (y >> 15) & 1;
  int b0  = y0 ^ y15;
  int b1  = y1 ^ b0;
  return x | b0 | (b1 << 1);
}

extern "C" __global__ __launch_bounds__(TPB)
void qsim16(const float* __restrict__ x,      // (512,16)
            const float* __restrict__ qw,     // (128,)
            const float* __restrict__ scale,  // (16,)
            const float* __restrict__ shift,  // (16,)
            float* __restrict__ out)          // (512,16)
{
  extern __shared__ float S[];               // 65536 floats = 256 KB
  const int b    = blockIdx.x;
  const int tid  = threadIdx.x;
  const int lane = tid & 31;
  // Wave index is uniform across the wave: pin it to an SGPR so the tile
  // loop below compiles as a uniform (s_cbranch) loop, not an EXEC-masked one.
  const int wave = __builtin_amdgcn_readfirstlane(tid >> 5);   // 0..31
  const int m    = lane & 15;                // WMMA A row / D col index
  const int hi   = lane >> 4;                // 0 = low half, 1 = high half
  const int koff = hi << 1;                  // A/B K interleave: 0 or 2
  const int ncol = lane & 15;                // B/D column

  // |0...0>
  #pragma unroll
  for (int j = 0; j < PER_T; ++j) S[tid * PER_T + j] = 0.0f;
  __syncthreads();
  if (tid == 0) S[0] = 1.0f;
  __syncthreads();

  for (int r = 0; r < 1 + NLAYERS; ++r) {
    // ---- RY round: nibble g = qubits 4g..4g+3 (qubit q <-> index bit q) ----
    #pragma unroll
    for (int g = 0; g < 4; ++g) {
      // cos/sin of half-angles for this nibble's 4 qubits
      float ch[4], sh4[4];
      #pragma unroll
      for (int i = 0; i < 4; ++i) {
        const int q = 4 * g + i;
        float th;
        if (r == 0) th = fmaf(x[b * NQ + q], scale[q], shift[q]); // encoding
        else        th = qw[(r - 1) * NQ + q];                    // layer weights
        __sincosf(0.5f * th, &sh4[i], &ch[i]);
      }
      // Build A fragments of U_g = RY3 (x) RY2 (x) RY1 (x) RY0 (16x16) in regs.
      // A layout (ISA 16x4 f32): lanes0-15 M=lane, VGPR0=K0/VGPR1=K1;
      //                          lanes16-31 M=lane-16, VGPR0=K2/VGPR1=K3.
      v2f A[4];
      #pragma unroll
      for (int kk = 0; kk < 4; ++kk) {
        #pragma unroll
        for (int c2 = 0; c2 < 2; ++c2) {
          const int k = 4 * kk + koff + c2;
          float e = 1.0f;
          #pragma unroll
          for (int i = 0; i < 4; ++i) {
            const int mb = (m >> i) & 1;
            const int kb = (k >> i) & 1;
            // RY(t) = [[c,-s],[s,c]] at t/2
            const float f = (mb == kb) ? ch[i] : (kb ? -sh4[i] : sh4[i]);
            e *= f;
          }
          A[kk][c2] = e;
        }
      }
      // 256 disjoint in-place 16x16 tiles; 8 per wave. Constant trip count +
      // scalar `wave` -> uniform control flow, EXEC stays all-ones at WMMA.
      #pragma unroll 2
      for (int tt = 0; tt < 8; ++tt) {
        const int t = (tt << 5) + wave;
        int base, kstride, nstride;
        if (g == 0)      { base = t * 256;                         kstride = 1;    nstride = 16; }
        else if (g == 1) { base = t * 256;                         kstride = 16;   nstride = 1;  }
        else if (g == 2) { base = (t >> 4) * 4096 + (t & 15) * 16; kstride = 256;  nstride = 1;  }
        else             { base = t * 16;                          kstride = 4096; nstride = 1;  }
        const int col = base + ncol * nstride;
        // B fragments: assumed symmetric layout (VGPR0 = K{0 low/2 high}, VGPR1 = K{1/3})
        v2f B[4];
        #pragma unroll
        for (int kk = 0; kk < 4; ++kk) {
          const int k0 = 4 * kk + koff;
          B[kk][0] = S[col + k0 * kstride];
          B[kk][1] = S[col + (k0 + 1) * kstride];
        }
        v8f acc = {0.f, 0.f, 0.f, 0.f, 0.f, 0.f, 0.f, 0.f};
        #pragma unroll
        for (int kk = 0; kk < 4; ++kk) {
          acc = __builtin_amdgcn_wmma_f32_16x16x4_f32(
              /*neg_a=*/false, A[kk], /*neg_b=*/false, B[kk],
              /*c_mod=*/(short)0, acc, /*reuse_a=*/false, /*reuse_b=*/false);
        }
        // D layout: VGPR rr -> row rr (low lanes) / rr+8 (high lanes), col=lane&15
        const int mb8 = hi * 8;
        #pragma unroll
        for (int rr = 0; rr < 8; ++rr)
          S[col + (mb8 + rr) * kstride] = acc[rr];
      }
      __syncthreads();
    }

    // ---- CNOT ring chain (layers only): one fused permutation ----
    if (r >= 1) {
      float stage[PER_T];
      #pragma unroll
      for (int j = 0; j < PER_T; ++j) {
        const int y = tid * PER_T + j;
        stage[j] = S[cnot_chain_src(y)];
      }
      __syncthreads();   // all gathers done before any overwrite
      #pragma unroll
      for (int j = 0; j < PER_T; ++j)
        S[tid * PER_T + j] = stage[j];
      __syncthreads();
    }
  }

  // ---- Measurement: <Z_q> = sum p*(1-2*bit_q) ----
  float accq[NQ];
  #pragma unroll
  for (int q = 0; q < NQ; ++q) accq[q] = 0.0f;
  #pragma unroll
  for (int j = 0; j < PER_T; ++j) {
    const int i = tid * PER_T + j;
    const float v = S[i];
    const float p = v * v;
    #pragma unroll
    for (int q = 0; q < NQ; ++q)
      accq[q] += ((i >> q) & 1) ? -p : p;
  }
  __syncthreads();                 // state dead; reuse LDS for reduction
  #pragma unroll
  for (int q = 0; q < NQ; ++q) S[q * TPB + tid] = accq[q];
  __syncthreads();
  for (int sh = 9; sh >= 0; --sh) {
    const int s = 1 << sh;
    for (int idx = tid; idx < NQ * s; idx += TPB) {
      const int q = idx >> sh;
      const int t = idx & (s - 1);
      S[q * TPB + t] += S[q * TPB + t + s];
    }
    __syncthreads();
  }
  if (tid < NQ) out[b * NQ + tid] = S[tid * TPB];
}

extern "C" void kernel_launch(void* const* d_in, const int* in_sizes, int n_in,
                              void* d_out, int out_size, void* d_ws, size_t ws_size,
                              hipStream_t stream) {
  (void)in_sizes; (void)n_in; (void)d_ws; (void)ws_size; (void)out_size;
  const float* x     = (const float*)d_in[0];
  const float* qw    = (const float*)d_in[1];
  const float* scale = (const float*)d_in[2];
  const float* shift = (const float*)d_in[3];
  float* out = (float*)d_out;

  const size_t lds_bytes = (size_t)NSTATE * sizeof(float);  // 256 KB (<= 320 KB WGP LDS)
  hipFuncSetAttribute((const void*)qsim16,
                      hipFuncAttributeMaxDynamicSharedMemorySize, (int)lds_bytes);
  hipLaunchKernelGGL(qsim16, dim3(512), dim3(TPB), lds_bytes, stream,
                     x, qw, scale, shift, out);
}